// Focus_37907381354554
// MI455X (gfx1250) — compile-verified
//
#include <hip/hip_runtime.h>
#include <hip/hip_bf16.h>

typedef float v2f __attribute__((ext_vector_type(2)));
typedef float v8f __attribute__((ext_vector_type(8)));

#define BN_EPS 1e-5f

// Problem dims (fixed by the reference)
constexpr int Bsz  = 16;
constexpr int Cin  = 3;
constexpr int H    = 640;
constexpr int Wd   = 640;
constexpr int Cout = 64;
constexpr int H2   = 320;
constexpr int W2   = 320;
constexpr int KC   = 12;                         // 4*Cin contraction dim
constexpr int TILES_PER_ROW = W2 / 16;           // 20 (exact -> tiles never cross rows)
constexpr int NTILES = Bsz * H2 * TILES_PER_ROW; // 102400 16-pixel tiles
constexpr int PLANE  = H2 * W2;                  // 102400 elements per output channel plane

// One wave computes a 64(M) x 16(N) output tile:
//   Y = (W * bn_scale) x P + bn_shift, then SiLU.
// 4 M-tiles x 3 K-steps = 12 x V_WMMA_F32_16X16X4_F32 per tile.
__global__ __launch_bounds__(256) void focus_wmma_kernel(
    const float* __restrict__ x, const float* __restrict__ Wt,
    const float* __restrict__ gamma, const float* __restrict__ beta,
    const float* __restrict__ rmean, const float* __restrict__ rvar,
    float* __restrict__ out)
{
    const int lane = threadIdx.x & 31;
    const int lo   = lane & 15;        // column within 16-lane group
    const int hi   = lane >> 4;        // which 16-lane half
    const int wave   = blockIdx.x * (blockDim.x >> 5) + (threadIdx.x >> 5);
    const int nwaves = gridDim.x * (blockDim.x >> 5);

    // ---- A fragments: W row pre-scaled by BN scale (folds the BN multiply) ----
    // A(16x4 f32) layout: lane = m + 16*(k>=2), vgpr = k&1  =>  this lane holds
    // m = lo, k = 2*hi + v  (v = component).  o = mt*16 + lo.
    float scale_m[4];
#pragma unroll
    for (int mt = 0; mt < 4; ++mt) {
        int o = mt * 16 + lo;
        scale_m[mt] = gamma[o] * rsqrtf(rvar[o] + BN_EPS);
    }
    v2f afrag[4][3];
#pragma unroll
    for (int mt = 0; mt < 4; ++mt) {
        int o = mt * 16 + lo;
#pragma unroll
        for (int kt = 0; kt < 3; ++kt) {
            int c0 = kt * 4 + hi * 2;
            afrag[mt][kt].x = Wt[o * KC + c0]     * scale_m[mt];
            afrag[mt][kt].y = Wt[o * KC + c0 + 1] * scale_m[mt];
        }
    }

    // ---- BN shift fragments, used to seed the WMMA accumulator ----
    // C/D(16x16 f32) layout: vgpr r, lanes<16 -> M=r, lanes>=16 -> M=r+8.
    v8f shiftfrag[4];
#pragma unroll
    for (int mt = 0; mt < 4; ++mt) {
#pragma unroll
        for (int r = 0; r < 8; ++r) {
            int o = mt * 16 + r + 8 * hi;
            float s = gamma[o] * rsqrtf(rvar[o] + BN_EPS);
            shiftfrag[mt][r] = beta[o] - rmean[o] * s;
        }
    }

    // ---- grid-stride over 16-pixel tiles (wave-uniform: EXEC stays all-1s) ----
    for (int t = wave; t < NTILES; t += nwaves) {
        int tw = t % TILES_PER_ROW;
        int h2 = (t / TILES_PER_ROW) % H2;
        int b  = t / (TILES_PER_ROW * H2);
        int w2 = tw * 16 + lo;

        // B fragments: P[c, n] space-to-depth gather straight from x.
        // B(4x16 f32) layout: lane = n + 16*(k>=2), vgpr = k&1  =>  this lane
        // holds n = lo, k = 2*hi + v; global channel c = 4*kt + k.
        // p channel c: group g = c/3 -> (dh,dw) = (g&1, g>>1), input ch ci = c%3.
        v2f bfrag[3];
#pragma unroll
        for (int kt = 0; kt < 3; ++kt) {
#pragma unroll
            for (int v = 0; v < 2; ++v) {
                int c  = kt * 4 + hi * 2 + v;
                int g  = c / 3;
                int ci = c % 3;
                int dh = g & 1;
                int dw = g >> 1;
                int idx = ((b * Cin + ci) * H + (2 * h2 + dh)) * Wd + (2 * w2 + dw);
                bfrag[kt][v] = x[idx];
            }
        }

        // Accumulate: acc = A0..2 * B0..2 + shift
        v8f acc[4];
#pragma unroll
        for (int mt = 0; mt < 4; ++mt) acc[mt] = shiftfrag[mt];
#pragma unroll
        for (int kt = 0; kt < 3; ++kt) {
#pragma unroll
            for (int mt = 0; mt < 4; ++mt) {
                acc[mt] = __builtin_amdgcn_wmma_f32_16x16x4_f32(
                    false, afrag[mt][kt], false, bfrag[kt],
                    (short)0, acc[mt], false, false);
            }
        }

        // SiLU + store.
        //  - sigmoid via single v_exp_f32 + v_rcp_f32 (no IEEE div sequence)
        //  - non-temporal stores: output (419MB) is write-once and exceeds L2,
        //    keep it from evicting the input stream
        //  - per-mt base pointer so the 8 r-stores become immediate-offset
        //    stores (r * 409600B fits the signed 24-bit IOFFSET)
#pragma unroll
        for (int mt = 0; mt < 4; ++mt) {
            float* outp = out + ((b * Cout + mt * 16 + 8 * hi) * H2 + h2) * W2 + w2;
#pragma unroll
            for (int r = 0; r < 8; ++r) {
                float y   = acc[mt][r];
                float sig = __builtin_amdgcn_rcpf(1.0f + __expf(-y));
                __builtin_nontemporal_store(y * sig, outp + r * PLANE);
            }
        }
    }
}

extern "C" void kernel_launch(void* const* d_in, const int* in_sizes, int n_in,
                              void* d_out, int out_size, void* d_ws, size_t ws_size,
                              hipStream_t stream) {
    const float* x     = (const float*)d_in[0];
    const float* Wt    = (const float*)d_in[1];
    const float* gamma = (const float*)d_in[2];
    const float* beta  = (const float*)d_in[3];
    const float* rmean = (const float*)d_in[4];
    const float* rvar  = (const float*)d_in[5];
    float* out = (float*)d_out;

    // 2048 blocks x 8 waves = 16384 waves; ~6.25 tiles/wave amortizes the
    // weight/BN fragment setup while saturating a memory-bound kernel.
    dim3 grid(2048), block(256);
    hipLaunchKernelGGL(focus_wmma_kernel, grid, block, 0, stream,
                       x, Wt, gamma, beta, rmean, rvar, out);
}